// ImitativeModel_53274774339711
// MI455X (gfx1250) — compile-verified
//
#include <hip/hip_runtime.h>
#include <hip/hip_bf16.h>
#include <math.h>

#define BATCH   16384
#define TSTEPS  4
#define NGOAL   4
#define NSTEPS  8
#define INV_B   (1.0f/16384.0f)
#define LOG2PI  1.8378770664093453f
#define LOGK    1.3862943611198906f

typedef __attribute__((ext_vector_type(2))) float v2f;
typedef __attribute__((ext_vector_type(8))) float v8f;

// One 16x16 f32 tile via V_WMMA_F32_16X16X4_F32, K-loop over LDS operands.
// A: 16 x K (row-major, stride lda). B: K x 16 (row-major, stride ldb).
// Fragment layout per CDNA5 ISA 7.12.2:
//   A: VGPR0 = {K=k, lanes0-15 | K=k+2, lanes16-31}, VGPR1 = {k+1 | k+3}, M = lane%16
//   B: symmetric, N = lane%16
//   C: VGPR v = row (v + 8*(lane>=16)), col lane%16
__device__ __forceinline__ v8f wmma_f32_tile(const float* A, int lda,
                                             const float* B, int ldb,
                                             int K, v8f c) {
  const int lane = threadIdx.x & 31;
  const int hf   = lane >> 4;
  const int idx  = lane & 15;
  for (int k = 0; k < K; k += 4) {
    v2f a, b;
    const int ka = k + 2 * hf;
    a.x = A[idx * lda + ka];
    a.y = A[idx * lda + ka + 1];
    b.x = B[ka * ldb + idx];
    b.y = B[(ka + 1) * ldb + idx];
    c = __builtin_amdgcn_wmma_f32_16x16x4_f32(false, a, false, b, (short)0, c, false, false);
  }
  return c;
}

__device__ __forceinline__ float sigm(float x) { return 1.0f / (1.0f + expf(-x)); }
__device__ __forceinline__ float softplusf(float x) { return (x > 20.0f) ? x : log1pf(expf(x)); }

// ------------------------------------------------------------------ init
__global__ void k_init(const float* __restrict__ base, float* __restrict__ x,
                       float* __restrict__ m, float* __restrict__ v,
                       float* __restrict__ xb, float* __restrict__ scal) {
  int i = blockIdx.x * blockDim.x + threadIdx.x;
  if (i < BATCH * 8) {
    float b = base[i & 7];
    x[i] = b; xb[i] = b; m[i] = 0.0f; v[i] = 0.0f;
  }
  if (i == 0) { scal[0] = 0.0f; scal[1] = 1000.0f; scal[2] = 0.0f; }
}

// ------------------------------------------------------------------ context MLP
#define MLP_WAVES 4
#define MW1 0
#define MW2 (MW1 + 136*68)
#define MW3 (MW2 + 64*68)
#define MB1 (MW3 + 64*68)
#define MB2 (MB1 + 64)
#define MB3 (MB2 + 64)
#define MPW (MB3 + 64)
#define MPW_FEAT 0
#define MPW_ZA   (16*140)
#define MPW_ZB   (MPW_ZA + 16*68)
#define MPW_SIZE (MPW_ZB + 16*68)
#define MLP_LDS_FLOATS (MPW + MLP_WAVES*MPW_SIZE)

__global__ void k_mlp(const float* __restrict__ vf, const float* __restrict__ vel,
                      const float* __restrict__ tl, const float* __restrict__ tls,
                      const float* __restrict__ Wm1, const float* __restrict__ bm1,
                      const float* __restrict__ Wm2, const float* __restrict__ bm2,
                      const float* __restrict__ Wm3, const float* __restrict__ bm3,
                      float* __restrict__ z_ctx) {
  extern __shared__ float lds[];
  const int tid  = threadIdx.x;
  const int wave = tid >> 5;
  const int lane = tid & 31;
  const int hf   = lane >> 4;
  const int idx  = lane & 15;

  for (int i = tid; i < 136 * 64; i += blockDim.x) {
    int k = i >> 6, n = i & 63;
    lds[MW1 + k * 68 + n] = (k < 133) ? Wm1[k * 64 + n] : 0.0f;
  }
  for (int i = tid; i < 64 * 64; i += blockDim.x) {
    int k = i >> 6, n = i & 63;
    lds[MW2 + k * 68 + n] = Wm2[i];
    lds[MW3 + k * 68 + n] = Wm3[i];
  }
  for (int i = tid; i < 64; i += blockDim.x) {
    lds[MB1 + i] = bm1[i]; lds[MB2 + i] = bm2[i]; lds[MB3 + i] = bm3[i];
  }

  float* feat = lds + MPW + wave * MPW_SIZE + MPW_FEAT;
  float* zA   = lds + MPW + wave * MPW_SIZE + MPW_ZA;
  float* zB   = lds + MPW + wave * MPW_SIZE + MPW_ZB;
  const int row0 = blockIdx.x * (MLP_WAVES * 16) + wave * 16;

  for (int i = lane; i < 16 * 136; i += 32) {
    int r = i / 136, c = i % 136;
    int gr = row0 + r;
    float val;
    if (c < 128)      val = vf[gr * 128 + c];
    else if (c < 131) val = vel[gr * 3 + (c - 128)];
    else if (c == 131) val = tl[gr];
    else if (c == 132) val = tls[gr];
    else               val = 0.0f;
    feat[r * 140 + c] = val;
  }
  __syncthreads();

  for (int nt = 0; nt < 4; nt++) {
    v8f c = {};
    c = wmma_f32_tile(feat, 140, lds + MW1 + nt * 16, 68, 136, c);
#pragma unroll
    for (int vv = 0; vv < 8; vv++) {
      int r = vv + 8 * hf;
      zA[r * 68 + nt * 16 + idx] = fmaxf(c[vv] + lds[MB1 + nt * 16 + idx], 0.0f);
    }
  }
  __syncthreads();
  for (int nt = 0; nt < 4; nt++) {
    v8f c = {};
    c = wmma_f32_tile(zA, 68, lds + MW2 + nt * 16, 68, 64, c);
#pragma unroll
    for (int vv = 0; vv < 8; vv++) {
      int r = vv + 8 * hf;
      zB[r * 68 + nt * 16 + idx] = fmaxf(c[vv] + lds[MB2 + nt * 16 + idx], 0.0f);
    }
  }
  __syncthreads();
  for (int nt = 0; nt < 4; nt++) {
    v8f c = {};
    c = wmma_f32_tile(zB, 68, lds + MW3 + nt * 16, 68, 64, c);
#pragma unroll
    for (int vv = 0; vv < 8; vv++) {
      int r = vv + 8 * hf;
      z_ctx[(row0 + r) * 64 + nt * 16 + idx] = fmaxf(c[vv] + lds[MB3 + nt * 16 + idx], 0.0f);
    }
  }
}

// ------------------------------------------------------------------ flow fwd + bwd
#define FB_WAVES 2
#define FWHHT 0
#define FWHH  (FWHHT + 64*196)
#define FWIH  (FWHH + 192*68)
#define FWL1  (FWIH + 384)
#define FWL2  (FWL1 + 64*36)
#define FBIH  (FWL2 + 128)
#define FBHH  (FBIH + 192)
#define FBL1  (FBHH + 192)
#define FBL2  (FBL1 + 32)
#define FPW   (FBL2 + 4)
#define PW_HH  0
#define PW_YH  (PW_HH + 5*16*68)
#define PW_SCH (PW_YH + 5*16*2)
#define PW_GHB (PW_SCH + 4*16*2)
#define PW_AB  (PW_GHB + 16*196)
#define PW_DLS (PW_AB + 16*36)
#define PW_DAP (PW_DLS + 16*4)
#define PW_DHC (PW_DAP + 16*36)
#define PW_DY  (PW_DHC + 16*68)
#define PW_MSC (PW_DY + 16*2)
#define PW_SIZE (PW_MSC + 16*8)
#define FB_LDS_FLOATS (FPW + FB_WAVES*PW_SIZE)

__global__ void k_fwdbwd(const float* __restrict__ xin, const float* __restrict__ goal,
                         const float* __restrict__ z_ctx,
                         const float* __restrict__ w_ih, const float* __restrict__ w_hh,
                         const float* __restrict__ b_ih, const float* __restrict__ b_hh,
                         const float* __restrict__ Wl1, const float* __restrict__ bl1,
                         const float* __restrict__ Wl2, const float* __restrict__ bl2,
                         float* __restrict__ grad, float* __restrict__ scal,
                         float* __restrict__ yout, int final_mode) {
  extern __shared__ float lds[];
  const int tid  = threadIdx.x;
  const int wave = tid >> 5;
  const int lane = tid & 31;
  const int hf   = lane >> 4;
  const int idx  = lane & 15;
  const int rr   = idx;  // batch row within the wave tile (elementwise phases)

  // ---- stage shared weights (w_hh both orders for fwd/bwd GEMMs)
  for (int i = tid; i < 192 * 64; i += blockDim.x) {
    int j = i >> 6, k = i & 63;
    float w = w_hh[i];
    lds[FWHH  + j * 68  + k] = w;   // (192 x 64)  B for  d_gh @ w_hh
    lds[FWHHT + k * 196 + j] = w;   // (64 x 192)  B for  h @ w_hh^T
  }
  for (int i = tid; i < 384; i += blockDim.x) lds[FWIH + i] = w_ih[i];
  for (int i = tid; i < 64 * 32; i += blockDim.x) {
    int k = i >> 5, m = i & 31;
    lds[FWL1 + k * 36 + m] = Wl1[i];
  }
  for (int i = tid; i < 128; i += blockDim.x) lds[FWL2 + i] = Wl2[i];
  for (int i = tid; i < 192; i += blockDim.x) { lds[FBIH + i] = b_ih[i]; lds[FBHH + i] = b_hh[i]; }
  for (int i = tid; i < 32; i += blockDim.x) lds[FBL1 + i] = bl1[i];
  for (int i = tid; i < 4; i += blockDim.x)  lds[FBL2 + i] = bl2[i];

  float* pw  = lds + FPW + wave * PW_SIZE;
  float* hh  = pw + PW_HH;    // h_{-1..3}: [5][16][68]
  float* yh  = pw + PW_YH;    // y_{-1..3}: [5][16][2]
  float* sch = pw + PW_SCH;   // scales:    [4][16][2]
  float* ghb = pw + PW_GHB;   // gh / d_gh: [16][196]
  float* ab  = pw + PW_AB;    // head act:  [16][36]
  float* dls = pw + PW_DLS;   // [16][4]
  float* dap = pw + PW_DAP;   // [16][36]
  float* dhc = pw + PW_DHC;   // dL/dh carry [16][68]
  float* dYb = pw + PW_DY;    // dL/dy carry [16][2]
  float* msc = pw + PW_MSC;   // misc comm   [16][8]

  const int rowbase = blockIdx.x * (FB_WAVES * 16) + wave * 16;
  const int grow = rowbase + rr;

  for (int i = lane; i < 16 * 64; i += 32) {
    int r = i >> 6, k = i & 63;
    hh[r * 68 + k] = z_ctx[(rowbase + r) * 64 + k];
  }
  for (int i = lane; i < 32; i += 32) yh[i] = 0.0f;
  float acc_ls = 0.0f, acc_x2 = 0.0f;
  __syncthreads();

  // ======================= forward =======================
  for (int t = 0; t < TSTEPS; t++) {
    for (int nt = 0; nt < 12; nt++) {          // gh = h_prev @ w_hh^T + b_hh
      v8f c = {};
      c = wmma_f32_tile(hh + t * (16 * 68), 68, lds + FWHHT + nt * 16, 196, 64, c);
#pragma unroll
      for (int vv = 0; vv < 8; vv++) {
        int r = vv + 8 * hf;
        ghb[r * 196 + nt * 16 + idx] = c[vv] + lds[FBHH + nt * 16 + idx];
      }
    }
    __syncthreads();
    {                                          // GRU gates, 32 lanes x 32 j each
      float y0 = yh[t * 32 + rr * 2 + 0];
      float y1 = yh[t * 32 + rr * 2 + 1];
      for (int jj = 0; jj < 32; jj++) {
        int j = 32 * hf + jj;
        float gir = lds[FWIH + j*2]*y0       + lds[FWIH + j*2+1]*y1       + lds[FBIH + j];
        float giu = lds[FWIH + (64+j)*2]*y0  + lds[FWIH + (64+j)*2+1]*y1  + lds[FBIH + 64+j];
        float gin = lds[FWIH + (128+j)*2]*y0 + lds[FWIH + (128+j)*2+1]*y1 + lds[FBIH + 128+j];
        float rg = sigm(gir + ghb[rr*196 + j]);
        float ug = sigm(giu + ghb[rr*196 + 64 + j]);
        float ng = tanhf(gin + rg * ghb[rr*196 + 128 + j]);
        float hp = hh[t*(16*68) + rr*68 + j];
        hh[(t+1)*(16*68) + rr*68 + j] = (1.0f - ug) * ng + ug * hp;
      }
    }
    __syncthreads();
    for (int mm = 0; mm < 16; mm++) {          // head: a = relu(h @ Wl1 + bl1)
      int m = 16 * hf + mm;
      float s = lds[FBL1 + m];
      for (int k = 0; k < 64; k++) s += hh[(t+1)*(16*68) + rr*68 + k] * lds[FWL1 + k*36 + m];
      ab[rr * 36 + m] = fmaxf(s, 0.0f);
    }
    __syncthreads();
    {                                          // ls = a @ Wl2 + bl2 -> dloc / scale
      int j0 = 2 * hf;
      float l0 = lds[FBL2 + j0], l1 = lds[FBL2 + j0 + 1];
      for (int m = 0; m < 32; m++) {
        float av = ab[rr * 36 + m];
        l0 += av * lds[FWL2 + m*4 + j0];
        l1 += av * lds[FWL2 + m*4 + j0 + 1];
      }
      if (hf == 0) { msc[rr*8 + 0] = l0; msc[rr*8 + 1] = l1; }
      else {
        sch[t*32 + rr*2 + 0] = softplusf(l0) + 0.001f;
        sch[t*32 + rr*2 + 1] = softplusf(l1) + 0.001f;
      }
    }
    __syncthreads();
    {                                          // y_t = y_{t-1} + dloc + scale*x_t
      int d = hf;
      float sc = sch[t*32 + rr*2 + d];
      float xv = xin[grow*8 + t*2 + d];
      yh[(t+1)*32 + rr*2 + d] = yh[t*32 + rr*2 + d] + msc[rr*8 + d] + sc * xv;
      acc_ls += logf(sc);
      acc_x2 += 0.5f * xv * xv;
    }
    __syncthreads();
  }

  if (final_mode) {
    int d = hf;
    for (int t = 0; t < TSTEPS; t++)
      yout[grow*8 + t*2 + d] = yh[(t+1)*32 + rr*2 + d];
    return;
  }

  // ======================= loss + goal logsumexp =======================
  msc[rr*8 + 2 + hf] = acc_ls;
  msc[rr*8 + 4 + hf] = acc_x2;
  for (int i = lane; i < 16 * 68; i += 32) dhc[i] = 0.0f;
  __syncthreads();
  if (hf == 0) {
    float yl0 = yh[4*32 + rr*2 + 0];
    float yl1 = yh[4*32 + rr*2 + 1];
    float comp[NGOAL];
    float mx = -1e30f;
    for (int k = 0; k < NGOAL; k++) {
      float g0 = goal[grow*8 + k*2 + 0];
      float g1 = goal[grow*8 + k*2 + 1];
      float d0 = yl0 - g0, d1 = yl1 - g1;
      comp[k] = -0.5f * (d0*d0 + d1*d1) - LOG2PI;
      mx = fmaxf(mx, comp[k]);
    }
    float se = 0.0f, yb0 = 0.0f, yb1 = 0.0f;
    for (int k = 0; k < NGOAL; k++) {
      float w = expf(comp[k] - mx);
      se += w;
      yb0 += w * goal[grow*8 + k*2 + 0];
      yb1 += w * goal[grow*8 + k*2 + 1];
    }
    float lp = mx + logf(se) - LOGK;
    yb0 /= se; yb1 /= se;
    float logabsdet = msc[rr*8+2] + msc[rr*8+3];
    float halfsumx2 = msc[rr*8+4] + msc[rr*8+5];
    float log_prob  = -halfsumx2 - 4.0f * LOG2PI;
    msc[rr*8 + 6] = log_prob - logabsdet + lp;   // loss = -mean(contrib)
    dYb[rr*2 + 0] = yl0 - yb0;                   // dL_elem/dy3
    dYb[rr*2 + 1] = yl1 - yb1;
  }
  __syncthreads();
  if (tid == 0) {
    float s = 0.0f;
    for (int w = 0; w < FB_WAVES; w++)
      for (int r = 0; r < 16; r++)
        s += lds[FPW + w * PW_SIZE + PW_MSC + r * 8 + 6];
    atomicAdd(&scal[0], s);
  }
  __syncthreads();

  // ======================= backward =======================
  for (int t = TSTEPS - 1; t >= 0; t--) {
    for (int nt = 0; nt < 12; nt++) {          // recompute gh for step t
      v8f c = {};
      c = wmma_f32_tile(hh + t * (16 * 68), 68, lds + FWHHT + nt * 16, 196, 64, c);
#pragma unroll
      for (int vv = 0; vv < 8; vv++) {
        int r = vv + 8 * hf;
        ghb[r * 196 + nt * 16 + idx] = c[vv] + lds[FBHH + nt * 16 + idx];
      }
    }
    __syncthreads();
    {                                          // d_ls and grad_x
      int d = hf;
      float sc = sch[t*32 + rr*2 + d];
      float xv = xin[grow*8 + t*2 + d];
      float dy = dYb[rr*2 + d];
      float dsc = dy * xv + 1.0f / sc;                 // from y_t and +log(scale)
      float sig = 1.0f - expf(-(sc - 0.001f));          // softplus'(s)
      dls[rr*4 + d]     = dy;                           // d dloc
      dls[rr*4 + 2 + d] = dsc * sig;                    // d s_raw
      grad[grow*8 + t*2 + d] = (xv + dy * sc) * INV_B;  // + direct 0.5*x^2 term
    }
    __syncthreads();
    for (int mm = 0; mm < 16; mm++) {          // head backward (recompute relu mask)
      int m = 16 * hf + mm;
      float da = 0.0f;
      for (int j = 0; j < 4; j++) da += lds[FWL2 + m*4 + j] * dls[rr*4 + j];
      float s = lds[FBL1 + m];
      for (int k = 0; k < 64; k++) s += hh[(t+1)*(16*68) + rr*68 + k] * lds[FWL1 + k*36 + m];
      dap[rr*36 + m] = (s > 0.0f) ? da : 0.0f;
    }
    __syncthreads();
    float dHreg[32];                           // dL/dh_t  (k = 32*hf + kk)
    for (int kk = 0; kk < 32; kk++) {
      int k = 32 * hf + kk;
      float dh = dhc[rr*68 + k];
      for (int m = 0; m < 32; m++) dh += lds[FWL1 + k*36 + m] * dap[rr*36 + m];
      dHreg[kk] = dh;
    }
    {                                          // gate backward; write d_gh in place
      float y0 = yh[t*32 + rr*2 + 0];
      float y1 = yh[t*32 + rr*2 + 1];
      float dyp0 = 0.0f, dyp1 = 0.0f;
      for (int jj = 0; jj < 32; jj++) {
        int j = 32 * hf + jj;
        float gir = lds[FWIH + j*2]*y0       + lds[FWIH + j*2+1]*y1       + lds[FBIH + j];
        float giu = lds[FWIH + (64+j)*2]*y0  + lds[FWIH + (64+j)*2+1]*y1  + lds[FBIH + 64+j];
        float gin = lds[FWIH + (128+j)*2]*y0 + lds[FWIH + (128+j)*2+1]*y1 + lds[FBIH + 128+j];
        float ghr = ghb[rr*196 + j];
        float ghu = ghb[rr*196 + 64 + j];
        float ghn = ghb[rr*196 + 128 + j];
        float rg = sigm(gir + ghr);
        float ug = sigm(giu + ghu);
        float ng = tanhf(gin + rg * ghn);
        float hp = hh[t*(16*68) + rr*68 + j];
        float dH = dHreg[jj];
        float dn = dH * (1.0f - ug);
        float du = dH * (hp - ng);
        float dhp = dH * ug;
        float dpn = dn * (1.0f - ng * ng);
        float dr  = dpn * ghn;
        float dghn = dpn * rg;
        float dpr = dr * rg * (1.0f - rg);
        float dpu = du * ug * (1.0f - ug);
        ghb[rr*196 + j]        = dpr;
        ghb[rr*196 + 64 + j]   = dpu;
        ghb[rr*196 + 128 + j]  = dghn;
        dhc[rr*68 + j] = dhp;
        dyp0 += lds[FWIH + j*2]*dpr     + lds[FWIH + (64+j)*2]*dpu     + lds[FWIH + (128+j)*2]*dpn;
        dyp1 += lds[FWIH + j*2+1]*dpr   + lds[FWIH + (64+j)*2+1]*dpu   + lds[FWIH + (128+j)*2+1]*dpn;
      }
      msc[rr*8 + hf*2 + 0] = dyp0;
      msc[rr*8 + hf*2 + 1] = dyp1;
    }
    __syncthreads();
    for (int nt = 0; nt < 4; nt++) {           // d_h_prev += d_gh @ w_hh
      v8f c = {};
      c = wmma_f32_tile(ghb, 196, lds + FWHH + nt * 16, 68, 192, c);
#pragma unroll
      for (int vv = 0; vv < 8; vv++) {
        int r = vv + 8 * hf;
        dhc[r * 68 + nt * 16 + idx] += c[vv];
      }
    }
    {                                          // dY_{t-1} = dY_t + w_ih^T d_gi
      int d = hf;
      dYb[rr*2 + d] += msc[rr*8 + 0 + d] + msc[rr*8 + 2 + d];
    }
    __syncthreads();
  }
}

// ------------------------------------------------------------------ loss compare
__global__ void k_finalize(float* __restrict__ scal) {
  float loss = -scal[0] * INV_B;
  float better = (loss < scal[1]) ? 1.0f : 0.0f;
  if (better > 0.5f) scal[1] = loss;
  scal[2] = better;
  scal[0] = 0.0f;    // reset accumulator for next step
}

// ------------------------------------------------------------------ Adam update
__global__ void k_adam(float* __restrict__ x, float* __restrict__ m, float* __restrict__ v,
                       float* __restrict__ xb, const float* __restrict__ grad,
                       const float* __restrict__ scal, float bc1, float bc2) {
  int i = blockIdx.x * blockDim.x + threadIdx.x;
  if (i >= BATCH * 8) return;
  float g = grad[i];
  float mm = 0.9f * m[i] + 0.1f * g;
  float vv = 0.999f * v[i] + 0.001f * g * g;
  m[i] = mm; v[i] = vv;
  float mh = mm / bc1;
  float vh = vv / bc2;
  float xn = x[i] - 0.1f * mh / (sqrtf(vh) + 1e-8f);
  x[i] = xn;
  if (scal[2] > 0.5f) xb[i] = xn;
}

// ------------------------------------------------------------------ launch
extern "C" void kernel_launch(void* const* d_in, const int* in_sizes, int n_in,
                              void* d_out, int out_size, void* d_ws, size_t ws_size,
                              hipStream_t stream) {
  const float* vf   = (const float*)d_in[0];
  const float* vel  = (const float*)d_in[1];
  const float* tl   = (const float*)d_in[2];
  const float* tls  = (const float*)d_in[3];
  const float* goal = (const float*)d_in[4];
  const float* base = (const float*)d_in[5];
  const float* Wm1  = (const float*)d_in[6];
  const float* bm1  = (const float*)d_in[7];
  const float* Wm2  = (const float*)d_in[8];
  const float* bm2  = (const float*)d_in[9];
  const float* Wm3  = (const float*)d_in[10];
  const float* bm3  = (const float*)d_in[11];
  const float* w_ih = (const float*)d_in[12];
  const float* w_hh = (const float*)d_in[13];
  const float* b_ih = (const float*)d_in[14];
  const float* b_hh = (const float*)d_in[15];
  const float* Wl1  = (const float*)d_in[16];
  const float* bl1  = (const float*)d_in[17];
  const float* Wl2  = (const float*)d_in[18];
  const float* bl2  = (const float*)d_in[19];

  float* ws    = (float*)d_ws;
  float* z_ctx = ws;
  float* x     = z_ctx + (size_t)BATCH * 64;
  float* m     = x  + (size_t)BATCH * 8;
  float* v     = m  + (size_t)BATCH * 8;
  float* xb    = v  + (size_t)BATCH * 8;
  float* g     = xb + (size_t)BATCH * 8;
  float* scal  = g  + (size_t)BATCH * 8;
  float* yout  = (float*)d_out;

  k_init<<<(BATCH * 8 + 255) / 256, 256, 0, stream>>>(base, x, m, v, xb, scal);
  k_mlp<<<BATCH / (MLP_WAVES * 16), MLP_WAVES * 32, MLP_LDS_FLOATS * sizeof(float), stream>>>(
      vf, vel, tl, tls, Wm1, bm1, Wm2, bm2, Wm3, bm3, z_ctx);

  double b1t = 1.0, b2t = 1.0;
  for (int s = 1; s <= NSTEPS; s++) {
    k_fwdbwd<<<BATCH / (FB_WAVES * 16), FB_WAVES * 32, FB_LDS_FLOATS * sizeof(float), stream>>>(
        x, goal, z_ctx, w_ih, w_hh, b_ih, b_hh, Wl1, bl1, Wl2, bl2, g, scal, yout, 0);
    k_finalize<<<1, 1, 0, stream>>>(scal);
    b1t *= 0.9; b2t *= 0.999;
    k_adam<<<(BATCH * 8 + 255) / 256, 256, 0, stream>>>(
        x, m, v, xb, g, scal, (float)(1.0 - b1t), (float)(1.0 - b2t));
  }
  k_fwdbwd<<<BATCH / (FB_WAVES * 16), FB_WAVES * 32, FB_LDS_FLOATS * sizeof(float), stream>>>(
      xb, goal, z_ctx, w_ih, w_hh, b_ih, b_hh, Wl1, bl1, Wl2, bl2, g, scal, yout, 1);

  (void)in_sizes; (void)n_in; (void)out_size; (void)ws_size;
}